// GAT_Encoder_46875273069315
// MI455X (gfx1250) — compile-verified
//
#include <hip/hip_runtime.h>
#include <math.h>

// Problem constants (match reference)
#define NN 25000      // nodes
#define NE 400000     // edges
#define NF 256        // in features
#define NC 32         // hidden per head (NHID)
#define NH 12         // heads
#define HC 384        // NH*NC
#define SLOPE 0.2f
#define KTILE 64      // K chunk staged in LDS (both GEMMs have K % 64 == 0)

typedef float v2f __attribute__((ext_vector_type(2)));
typedef float v8f __attribute__((ext_vector_type(8)));

// ---------------------------------------------------------------------------
// order-preserving float<->uint encoding for atomicMax-based segment max
__device__ __forceinline__ unsigned enc_f32(float x) {
  unsigned u = __float_as_uint(x);
  return (u & 0x80000000u) ? ~u : (u | 0x80000000u);
}
__device__ __forceinline__ float dec_f32(unsigned e) {
  unsigned u = (e & 0x80000000u) ? (e ^ 0x80000000u) : ~e;
  return __uint_as_float(u);
}

// ---------------------------------------------------------------------------
// GEMM: C[M,N] = A[M,K] @ B[K,N], row-major. Block = 8 waves = 16 rows x 128
// cols of C; each wave owns one 16x16 tile (f32 WMMA, K-step 4). The 16xKTILE
// A panel is staged once per block into LDS via the CDNA5 async-to-LDS engine
// and shared by all 8 waves (8x reduction of A read traffic vs per-wave
// loads). Requires: K % KTILE == 0, Ncol % 128 == 0 (true for both layers).
__global__ __launch_bounds__(256)
void gemm_wmma(const float* __restrict__ A, const float* __restrict__ B,
               float* __restrict__ C, int M, int K, int Ncol) {
  __shared__ __align__(16) float As[16 * KTILE];   // 4 KB
  const int tid  = threadIdx.x;
  const int wave = tid >> 5;        // 0..7 -> N subtile
  const int lane = tid & 31;
  const int half = lane >> 4;       // 0: lanes 0-15, 1: lanes 16-31
  const int l    = lane & 15;
  const int row0 = blockIdx.y * 16;
  const int col0 = blockIdx.x * 128 + wave * 16;

  // staging: one b128 (4 floats) per thread: row sr (0..15), chunk sc (0..15)
  const int sr = tid >> 4;
  const int sc = tid & 15;
  int arow = row0 + sr;
  if (arow >= M) arow = M - 1;      // clamp: rows >= M only feed discarded C rows
  const unsigned lds_dst =
      (unsigned)(size_t)As + (unsigned)((sr * KTILE + sc * 4) * 4);

  v8f acc = {0.f, 0.f, 0.f, 0.f, 0.f, 0.f, 0.f, 0.f};
  for (int k0 = 0; k0 < K; k0 += KTILE) {
    __syncthreads();                // previous chunk fully consumed
    const unsigned long long gsrc =
        (unsigned long long)(A + (size_t)arow * K + k0 + sc * 4);
    asm volatile("global_load_async_to_lds_b128 %0, %1, off"
                 :: "v"(lds_dst), "v"(gsrc) : "memory");
    asm volatile("s_wait_asynccnt 0x0" ::: "memory");
    __syncthreads();                // all waves' panels visible
#pragma unroll
    for (int kk = 0; kk < KTILE; kk += 4) {
      const int ka = kk + half * 2; // K pair handled by this lane half
      v2f a = *(const v2f*)&As[l * KTILE + ka];      // ds_load_b64
      v2f b;
      b.x = B[(size_t)(k0 + ka) * Ncol + col0 + l];
      b.y = B[(size_t)(k0 + ka + 1) * Ncol + col0 + l];
      acc = __builtin_amdgcn_wmma_f32_16x16x4_f32(false, a, false, b,
                                                  (short)0, acc, false, false);
    }
  }
#pragma unroll
  for (int j = 0; j < 8; ++j) {
    int r = row0 + half * 8 + j;
    if (r < M) C[(size_t)r * Ncol + col0 + l] = acc[j];
  }
}

// ---------------------------------------------------------------------------
__global__ void zero_f32(float* p, size_t n) {
  size_t i = (size_t)blockIdx.x * blockDim.x + threadIdx.x;
  size_t st = (size_t)gridDim.x * blockDim.x;
  for (; i < n; i += st) p[i] = 0.f;
}

// Fold W_e1/att_edge1 into Me[f][h] = sum_c W_e[f, h*NC+c] * att_e[h, c]
__global__ void prep_me(const float* __restrict__ We, const float* __restrict__ atte,
                        float* __restrict__ Me) {
  int t = threadIdx.x;            // 0..383
  int f = t / NH, h = t % NH;
  float s = 0.f;
  for (int c = 0; c < NC; ++c)
    s += We[(size_t)f * HC + h * NC + c] * atte[h * NC + c];
  Me[f * NH + h] = s;
}

// per-node attention scalars: a_s[n,h], a_d[n,h] from hx[n, h*NC+c]
__global__ void node_att(const float* __restrict__ hx, const float* __restrict__ as_w,
                         const float* __restrict__ ad_w, float* __restrict__ a_s,
                         float* __restrict__ a_d) {
  int id = blockIdx.x * blockDim.x + threadIdx.x;
  if (id >= NN * NH) return;
  int n = id / NH, h = id % NH;
  const float* hp = hx + (size_t)n * HC + h * NC;
  float s = 0.f, d = 0.f;
  for (int c = 0; c < NC; ++c) {
    float v = hp[c];
    s += v * as_w[h * NC + c];
    d += v * ad_w[h * NC + c];
  }
  a_s[id] = s;
  a_d[id] = d;
}

// per-edge folded edge-attr term; also accumulate per-dst sums + counts for
// the self-loop "mean edge_attr" fill value (linearity of the fold).
__global__ void edge_feat(const float* __restrict__ ea, const int* __restrict__ ei,
                          const float* __restrict__ Me, float* __restrict__ eep,
                          float* __restrict__ eep_sum, float* __restrict__ cnt) {
  int k = blockIdx.x * blockDim.x + threadIdx.x;
  if (k >= NE) return;
  int d = ei[NE + k];
  float acc[NH];
#pragma unroll
  for (int h = 0; h < NH; ++h) acc[h] = 0.f;
  const float* ep = ea + (size_t)k * NC;
  for (int f = 0; f < NC; ++f) {
    float v = ep[f];
#pragma unroll
    for (int h = 0; h < NH; ++h) acc[h] += v * Me[f * NH + h];
  }
#pragma unroll
  for (int h = 0; h < NH; ++h) {
    eep[(size_t)k * NH + h] = acc[h];
    atomicAdd(&eep_sum[(size_t)d * NH + h], acc[h]);
  }
  atomicAdd(&cnt[d], 1.f);
}

// pass A: attention logits (leaky relu) + segment max via encoded atomicMax
__global__ void attn_logits(const int* __restrict__ ei, const float* __restrict__ a_s,
                            const float* __restrict__ a_d, const float* __restrict__ eep,
                            const float* __restrict__ eep_sum, const float* __restrict__ cnt,
                            float* __restrict__ ebuf, unsigned* __restrict__ mbuf) {
  int id = blockIdx.x * blockDim.x + threadIdx.x;
  if (id >= (NE + NN) * NH) return;
  int k = id / NH, h = id % NH;
  int d;
  float e;
  if (k < NE) {
    int s = ei[k];
    d = ei[NE + k];
    e = a_s[s * NH + h] + a_d[d * NH + h];
    if (eep) e += eep[(size_t)k * NH + h];
  } else {
    int n = k - NE;
    d = n;
    e = a_s[n * NH + h] + a_d[n * NH + h];
    if (eep_sum) e += eep_sum[(size_t)n * NH + h] / fmaxf(cnt[n], 1.f);
  }
  e = (e > 0.f) ? e : SLOPE * e;
  ebuf[id] = e;
  atomicMax(&mbuf[d * NH + h], enc_f32(e));
}

// pass B: exp(e - max) + segment sum
__global__ void attn_exp(const int* __restrict__ ei, const unsigned* __restrict__ mbuf,
                         float* __restrict__ ebuf, float* __restrict__ den) {
  int id = blockIdx.x * blockDim.x + threadIdx.x;
  if (id >= (NE + NN) * NH) return;
  int k = id / NH, h = id % NH;
  int d = (k < NE) ? ei[NE + k] : (k - NE);
  float m = dec_f32(mbuf[d * NH + h]);
  float ex = expf(ebuf[id] - m);
  ebuf[id] = ex;
  atomicAdd(&den[d * NH + h], ex);
}

// pass C: out[dst] += alpha * hx[src]; one block (384 threads) per edge,
// thread t = (h, c): contiguous 384-float gather of hx[src], scatter-add.
__global__ __launch_bounds__(HC)
void attn_agg(const int* __restrict__ ei, const float* __restrict__ ebuf,
              const float* __restrict__ den, const float* __restrict__ hx,
              float* __restrict__ out) {
  int k = blockIdx.x;
  int t = threadIdx.x;
  int h = t >> 5;
  int s, d;
  if (k < NE) { s = ei[k]; d = ei[NE + k]; } else { s = d = k - NE; }
  float alpha = ebuf[(size_t)k * NH + h] / den[(size_t)d * NH + h];
  atomicAdd(&out[(size_t)d * HC + t], alpha * hx[(size_t)s * HC + t]);
}

// bias + ELU in place (layer-1 epilogue)
__global__ void bias_elu(float* __restrict__ o, const float* __restrict__ b) {
  size_t id = (size_t)blockIdx.x * blockDim.x + threadIdx.x;
  if (id >= (size_t)NN * HC) return;
  float v = o[id] + b[id % HC];
  o[id] = (v > 0.f) ? v : expm1f(v);
}

// layer-2 epilogue: mean over heads + bias -> h2 (first half of d_out)
__global__ void head_mean(const float* __restrict__ o, const float* __restrict__ b,
                          float* __restrict__ h2) {
  int id = blockIdx.x * blockDim.x + threadIdx.x;
  if (id >= NN * NC) return;
  int n = id / NC, c = id % NC;
  float s = 0.f;
#pragma unroll
  for (int h = 0; h < NH; ++h) s += o[(size_t)n * HC + h * NC + c];
  h2[id] = s * (1.f / NH) + b[c];
}

// log_softmax over 32 classes: one wave32 per row, lane = class
__global__ void log_softmax_rows(const float* __restrict__ h2, float* __restrict__ lp) {
  int row = blockIdx.x * (blockDim.x >> 5) + (threadIdx.x >> 5);
  int lane = threadIdx.x & 31;
  if (row >= NN) return;
  float v = h2[row * NC + lane];
  float m = v;
#pragma unroll
  for (int o = 16; o > 0; o >>= 1) m = fmaxf(m, __shfl_xor(m, o, 32));
  float ex = expf(v - m);
  float s = ex;
#pragma unroll
  for (int o = 16; o > 0; o >>= 1) s += __shfl_xor(s, o, 32);
  lp[row * NC + lane] = v - m - logf(s);
}

// ---------------------------------------------------------------------------
extern "C" void kernel_launch(void* const* d_in, const int* in_sizes, int n_in,
                              void* d_out, int out_size, void* d_ws, size_t ws_size,
                              hipStream_t stream) {
  const float* x      = (const float*)d_in[0];
  const int*   ei     = (const int*)d_in[1];     // [2,E]: src then dst
  const float* ea     = (const float*)d_in[2];
  const float* W1     = (const float*)d_in[3];
  const float* atts1  = (const float*)d_in[4];
  const float* attd1  = (const float*)d_in[5];
  const float* We1    = (const float*)d_in[6];
  const float* atte1  = (const float*)d_in[7];
  const float* b1     = (const float*)d_in[8];
  const float* W2     = (const float*)d_in[9];
  const float* atts2  = (const float*)d_in[10];
  const float* attd2  = (const float*)d_in[11];
  const float* b2     = (const float*)d_in[12];

  float* h2  = (float*)d_out;                 // [NN, NC]
  float* lp  = (float*)d_out + (size_t)NN * NC;

  // workspace arena (~123 MB), reused across layers
  char* ws = (char*)d_ws;
  size_t off = 0;
  auto take = [&](size_t bytes) -> char* {
    char* p = ws + off;
    off += (bytes + 255) & ~(size_t)255;
    return p;
  };
  float*    hx      = (float*)take((size_t)NN * HC * 4);          // hx1 then hx2
  float*    outb    = (float*)take((size_t)NN * HC * 4);          // out1/h1 then out2
  float*    ebuf    = (float*)take((size_t)(NE + NN) * NH * 4);   // e then exp(e)
  float*    eep     = (float*)take((size_t)NE * NH * 4);          // layer-1 only
  float*    a_s     = (float*)take((size_t)NN * NH * 4);
  float*    a_d     = (float*)take((size_t)NN * NH * 4);
  unsigned* mbuf    = (unsigned*)take((size_t)NN * NH * 4);
  float*    den     = (float*)take((size_t)NN * NH * 4);
  float*    eep_sum = (float*)take((size_t)NN * NH * 4);
  float*    cnt     = (float*)take((size_t)NN * 4);
  float*    Me      = (float*)take((size_t)NC * NH * 4);
  (void)in_sizes; (void)n_in; (void)out_size; (void)ws_size;

  const int ZB = 2048, ZT = 256;
  const int attT = 256, attG = ((NE + NN) * NH + attT - 1) / attT;
  const dim3 gg(HC / 128, (NN + 15) / 16);    // GEMM grid: 3 x 1563 blocks
  const dim3 gb(256);

  // ---------------- layer 1 ----------------
  zero_f32<<<ZB, ZT, 0, stream>>>(eep_sum, (size_t)NN * NH);
  zero_f32<<<ZB, ZT, 0, stream>>>(cnt, (size_t)NN);
  zero_f32<<<ZB, ZT, 0, stream>>>((float*)mbuf, (size_t)NN * NH);
  zero_f32<<<ZB, ZT, 0, stream>>>(den, (size_t)NN * NH);
  zero_f32<<<ZB, ZT, 0, stream>>>(outb, (size_t)NN * HC);

  prep_me<<<1, NC * NH, 0, stream>>>(We1, atte1, Me);
  gemm_wmma<<<gg, gb, 0, stream>>>(x, W1, hx, NN, NF, HC);
  node_att<<<(NN * NH + 255) / 256, 256, 0, stream>>>(hx, atts1, attd1, a_s, a_d);
  edge_feat<<<(NE + 255) / 256, 256, 0, stream>>>(ea, ei, Me, eep, eep_sum, cnt);
  attn_logits<<<attG, attT, 0, stream>>>(ei, a_s, a_d, eep, eep_sum, cnt, ebuf, mbuf);
  attn_exp<<<attG, attT, 0, stream>>>(ei, mbuf, ebuf, den);
  attn_agg<<<NE + NN, HC, 0, stream>>>(ei, ebuf, den, hx, outb);
  bias_elu<<<((size_t)NN * HC + 255) / 256, 256, 0, stream>>>(outb, b1);

  // ---------------- layer 2 ----------------
  gemm_wmma<<<gg, gb, 0, stream>>>(outb, W2, hx, NN, HC, HC);   // hx2 = h1 @ W2
  zero_f32<<<ZB, ZT, 0, stream>>>((float*)mbuf, (size_t)NN * NH);
  zero_f32<<<ZB, ZT, 0, stream>>>(den, (size_t)NN * NH);
  zero_f32<<<ZB, ZT, 0, stream>>>(outb, (size_t)NN * HC);        // h1 dead -> out2
  node_att<<<(NN * NH + 255) / 256, 256, 0, stream>>>(hx, atts2, attd2, a_s, a_d);
  attn_logits<<<attG, attT, 0, stream>>>(ei, a_s, a_d, nullptr, nullptr, nullptr, ebuf, mbuf);
  attn_exp<<<attG, attT, 0, stream>>>(ei, mbuf, ebuf, den);
  attn_agg<<<NE + NN, HC, 0, stream>>>(ei, ebuf, den, hx, outb);
  head_mean<<<(NN * NC + 255) / 256, 256, 0, stream>>>(outb, b2, h2);
  log_softmax_rows<<<(NN + 7) / 8, 256, 0, stream>>>(h2, lp);
}